// ControlFlowScanCDist2_46308337386062
// MI455X (gfx1250) — compile-verified
//
#include <hip/hip_runtime.h>

// Self-cdist out[i][j] = sqrt(||x_i - x_j||^2), x: [2048 x 64] fp32.
// LDS-tiled Gram matrix via V_WMMA_F32_16X16X4_F32 (wave32).
// Block = 8 waves = 128x128 output tile; panels staged in LDS once,
// via GLOBAL_LOAD_ASYNC_TO_LDS_B128 when the toolchain exposes it.

typedef __attribute__((ext_vector_type(2))) float v2f;
typedef __attribute__((ext_vector_type(8))) float v8f;
typedef __attribute__((ext_vector_type(4))) int   v4i;

#define NROWS 2048
#define DIM   64
#define TILE  128          // block output tile edge (rows and cols)
#define LSTR  68           // padded LDS row stride (floats): 68 % 64 = 4 -> conflict-free b64 reads

#if defined(__HIP_DEVICE_COMPILE__) && defined(__has_builtin)
#if __has_builtin(__builtin_amdgcn_global_load_async_to_lds_b128)
#define HAVE_ASYNC_LDS 1
#endif
#endif

__global__ __launch_bounds__(256)
void cdist2_wmma_kernel(const float* __restrict__ x, float* __restrict__ out) {
    __shared__ float As[TILE * LSTR];   // 34816 B
    __shared__ float Bs[TILE * LSTR];   // 34816 B

    const int tid  = threadIdx.x;
    const int lane = tid & 31;
    const int wave = tid >> 5;          // 0..7
    const int lo   = lane & 15;
    const int hi   = lane >> 4;         // 0 or 1

    const int rowPanel = blockIdx.y * TILE;   // first row of A panel
    const int colPanel = blockIdx.x * TILE;   // first row of B panel

    // ---- stage panels: 128 rows x 64 floats each, 16B per lane per op ----
    // lanes 0..15 cover 256B contiguous global segments.
#ifdef HAVE_ASYNC_LDS
    {
        typedef __attribute__((address_space(1))) v4i* gptr_t;  // global int4*
        typedef __attribute__((address_space(3))) v4i* lptr_t;  // LDS int4*
#pragma unroll
        for (int i = 0; i < 8; ++i) {
            const int idx = tid + i * 256;       // 0..2047
            const int r   = idx >> 4;            // row 0..127
            const int f   = idx & 15;            // float4 index 0..15
            __builtin_amdgcn_global_load_async_to_lds_b128(
                (gptr_t)(x + (size_t)(rowPanel + r) * DIM + 4 * f),
                (lptr_t)(As + r * LSTR + 4 * f), 0, 0);
            __builtin_amdgcn_global_load_async_to_lds_b128(
                (gptr_t)(x + (size_t)(colPanel + r) * DIM + 4 * f),
                (lptr_t)(Bs + r * LSTR + 4 * f), 0, 0);
        }
#if __has_builtin(__builtin_amdgcn_s_wait_asynccnt)
        __builtin_amdgcn_s_wait_asynccnt(0);
#else
        asm volatile("s_wait_asynccnt 0x0" ::: "memory");
#endif
    }
#else
#pragma unroll
    for (int i = 0; i < 8; ++i) {
        const int idx = tid + i * 256;
        const int r   = idx >> 4;
        const int f   = idx & 15;
        float4 va = *(const float4*)(x + (size_t)(rowPanel + r) * DIM + 4 * f);
        float4 vb = *(const float4*)(x + (size_t)(colPanel + r) * DIM + 4 * f);
        *(float4*)(As + r * LSTR + 4 * f) = va;
        *(float4*)(Bs + r * LSTR + 4 * f) = vb;
    }
#endif
    __syncthreads();

    // ---- 16 rows (this wave) x 128 cols (8 col tiles), K = 64 -> 16 wmma steps ----
    // A fragment: lane holds M = lo, VGPR v holds K = 4*kk + 2*hi + v (32-bit A 16x4 layout)
    // B fragment: mirrored layout, B[k][n] = Xcol[n][k] -> identical indexing per col tile.
    const float* aP = As + (wave * 16 + lo) * LSTR + 2 * hi;
    const float* bP = Bs + lo * LSTR + 2 * hi;

    v8f acc[8] = {};
    float na = 0.f;            // half-norm of A row (rowPanel + 16*wave + lo)
    float nb[8] = {};          // half-norms of B rows (colPanel + 16*ct + lo)

#pragma unroll 4
    for (int kk = 0; kk < 16; ++kk) {
        v2f a = *(const v2f*)(aP + 4 * kk);
        na += a.x * a.x + a.y * a.y;
#pragma unroll
        for (int ct = 0; ct < 8; ++ct) {
            v2f b = *(const v2f*)(bP + ct * (16 * LSTR) + 4 * kk);
            nb[ct] += b.x * b.x + b.y * b.y;
            acc[ct] = __builtin_amdgcn_wmma_f32_16x16x4_f32(
                false, a, false, b, (short)0, acc[ct], false, false);
        }
    }

    // ---- merge half-norms: lanes (lo) and (lo+16) each hold half of row lo's norm ----
    float naf = na + __shfl_xor(na, 16);         // lane L: full norm of A row (16*wave + (L&15))
    float njv[8];
#pragma unroll
    for (int ct = 0; ct < 8; ++ct)
        njv[ct] = nb[ct] + __shfl_xor(nb[ct], 16);  // lane L: norm of B row (16*ct + (L&15)) == its own N

    // ---- d = sqrt(max(ni + nj - 2*g, 0)); C/D layout: VGPR r, lane -> M = r + 8*hi, N = lo ----
#pragma unroll
    for (int r = 0; r < 8; ++r) {
        const int m  = r + 8 * hi;
        float ni = __shfl(naf, m);               // full norm of output row rowPanel+16*wave+m
        const size_t rowOff = (size_t)(rowPanel + wave * 16 + m) * NROWS + colPanel;
#pragma unroll
        for (int ct = 0; ct < 8; ++ct) {
            float d2 = ni + njv[ct] - 2.0f * acc[ct][r];
            d2 = d2 > 0.0f ? d2 : 0.0f;
            out[rowOff + ct * 16 + lo] = sqrtf(d2);
        }
    }
}

extern "C" void kernel_launch(void* const* d_in, const int* in_sizes, int n_in,
                              void* d_out, int out_size, void* d_ws, size_t ws_size,
                              hipStream_t stream) {
    (void)in_sizes; (void)n_in; (void)d_ws; (void)ws_size; (void)out_size;
    const float* x = (const float*)d_in[0];
    float* out = (float*)d_out;

    dim3 grid(NROWS / TILE, NROWS / TILE, 1);   // (16, 16)
    dim3 block(256, 1, 1);
    cdist2_wmma_kernel<<<grid, block, 0, stream>>>(x, out);
}